// BaseModel_69355131896059
// MI455X (gfx1250) — compile-verified
//
#include <hip/hip_runtime.h>

typedef float v2f __attribute__((ext_vector_type(2)));
typedef float v8f __attribute__((ext_vector_type(8)));

#define D_DIM 256

// ---------------------------------------------------------------------------
// Kernel 0: zero the workspace accumulators (acc[E*256] ++ cnt[E]).
// ---------------------------------------------------------------------------
__global__ void zero_ws_kernel(float* __restrict__ ws, int n) {
  int i = blockIdx.x * blockDim.x + threadIdx.x;
  if (i < n) ws[i] = 0.0f;
}

// ---------------------------------------------------------------------------
// Kernel 1: one wave32 per mention (ragged mean over <=16 token rows).
//
//   V_WMMA_F32_16X16X4_F32 as an exact-f32 masked reduction engine:
//     B (4x16): 4 gathered token rows x 16 dims (natural lane layout:
//               lane<16 -> K rows {k0,k0+1}, lane>=16 -> {k0+2,k0+3})
//     A (16x4): broadcast 0/1 validity mask, all 16 rows identical
//     D = A*B + C  ->  every row of the 16x16 tile = masked token sum for
//               the 16-dim slice; C accumulates the 4 K-chunks (K=16 tokens).
//
//   Masked-out rows are clamped to row len-1, so their 64B fetches are L0
//   hits on already-fetched lines: HBM traffic == exactly len rows/mention.
//
//   Two N-slices per iteration: lanes 0-15 read slice-A sums from ca[0]
//   (row M=0), lanes 16-31 read slice-B sums from cb[0] (row M=8, identical
//   values) -> full-wave global_atomic_add_f32 into the entity row.
// ---------------------------------------------------------------------------
__global__ __launch_bounds__(256) void mention_accum_kernel(
    const float* __restrict__ enc, const int* __restrict__ infoI,
    float* __restrict__ acc, float* __restrict__ cnt, int M) {
  int wave = blockIdx.x * (blockDim.x >> 5) + (threadIdx.x >> 5);
  if (wave >= M) return;  // wave-uniform: EXEC stays all-ones for WMMA
  int lane = threadIdx.x & 31;

  // info row: int64 {eid, etype, start, end}; values fit in low 32 bits.
  int eid   = __builtin_amdgcn_readfirstlane(infoI[8 * wave + 0]);
  int start = __builtin_amdgcn_readfirstlane(infoI[8 * wave + 4]);
  int end   = __builtin_amdgcn_readfirstlane(infoI[8 * wave + 6]);
  int len = end - start;                 // in [1, 16]
  float invLen = 1.0f / (float)len;

  int half  = lane >> 4;                 // which K-row pair this lane feeds
  int nlane = lane & 15;

  const float* base = enc + (size_t)start * D_DIM;
  float* accRow = acc + (size_t)eid * D_DIM;

  // Hoisted per-mention state: clamped row pointers + A-matrix masks.
  const float* p0[4];
  const float* p1[4];
  v2f amask[4];
#pragma unroll
  for (int kc = 0; kc < 4; ++kc) {
    int k0 = kc * 4 + half * 2;
    int k1 = k0 + 1;
    int r0 = (k0 < len) ? k0 : (len - 1);  // clamp: dup reads hit L0
    int r1 = (k1 < len) ? k1 : (len - 1);
    p0[kc] = base + (size_t)r0 * D_DIM;
    p1[kc] = base + (size_t)r1 * D_DIM;
    amask[kc].x = (k0 < len) ? 1.0f : 0.0f;
    amask[kc].y = (k1 < len) ? 1.0f : 0.0f;
  }

  for (int n2 = 0; n2 < 8; ++n2) {       // 8 x (two 16-dim slices) = 256 dims
    int colA = (n2 * 2) * 16 + nlane;
    int colB = colA + 16;
    v8f ca = {}, cb = {};
#pragma unroll
    for (int kc = 0; kc < 4; ++kc) {
      v2f ba, bb;
      ba.x = p0[kc][colA];
      ba.y = p1[kc][colA];
      bb.x = p0[kc][colB];
      bb.y = p1[kc][colB];
      ca = __builtin_amdgcn_wmma_f32_16x16x4_f32(false, amask[kc], false, ba,
                                                 (short)0, ca, false, false);
      cb = __builtin_amdgcn_wmma_f32_16x16x4_f32(false, amask[kc], false, bb,
                                                 (short)0, cb, false, false);
    }
    // ca[0]: lanes 0-15 = slice-A sums (row M=0).
    // cb[0]: lanes 16-31 = slice-B sums (row M=8, all rows identical).
    float v = (half ? cb[0] : ca[0]) * invLen;
    int myCol = (n2 * 2 + half) * 16 + nlane;
    atomicAdd(&accRow[myCol], v);        // full-wave atomic, no idle lanes
  }
  if (lane == 0) atomicAdd(&cnt[eid], 1.0f);
}

// ---------------------------------------------------------------------------
// Kernel 2: entities = acc / max(cnt, 1)
// ---------------------------------------------------------------------------
__global__ void finalize_kernel(const float* __restrict__ acc,
                                const float* __restrict__ cnt,
                                float* __restrict__ out, int n) {
  int i = blockIdx.x * blockDim.x + threadIdx.x;
  if (i < n) {
    int e = i >> 8;  // D_DIM == 256
    out[i] = acc[i] / fmaxf(cnt[e], 1.0f);
  }
}

extern "C" void kernel_launch(void* const* d_in, const int* in_sizes, int n_in,
                              void* d_out, int out_size, void* d_ws, size_t ws_size,
                              hipStream_t stream) {
  const float* enc = (const float*)d_in[0];       // (T, 256) f32
  const int* infoI = (const int*)d_in[1];         // (M, 4) int64 -> int32 pairs
  (void)n_in; (void)ws_size;

  int M = in_sizes[1] / 4;                        // 20000
  int E = out_size / D_DIM;                       // 4000

  float* acc = (float*)d_ws;                      // E * 256 accumulators
  float* cnt = acc + (size_t)E * D_DIM;           // E counts

  int nws = E * D_DIM + E;
  zero_ws_kernel<<<(nws + 255) / 256, 256, 0, stream>>>((float*)d_ws, nws);

  int wavesPerBlock = 8;                          // 256 threads = 8 wave32
  int blocks = (M + wavesPerBlock - 1) / wavesPerBlock;
  mention_accum_kernel<<<blocks, 256, 0, stream>>>(enc, infoI, acc, cnt, M);

  finalize_kernel<<<(out_size + 255) / 256, 256, 0, stream>>>(
      acc, cnt, (float*)d_out, out_size);
}